// GMMConv_31138512896561
// MI455X (gfx1250) — compile-verified
//
#include <hip/hip_runtime.h>
#include <hip/hip_bf16.h>

typedef __attribute__((ext_vector_type(2))) float v2f;
typedef __attribute__((ext_vector_type(8))) float v8f;
typedef __attribute__((address_space(3))) float lds_f;

#define D_FEAT 64
#define K_KERN 4

// ---------------------------------------------------------------------------
// Zero-fill scratch (deg, agg).
// ---------------------------------------------------------------------------
__global__ void gmm_zero_kernel(float* __restrict__ p, size_t n) {
    size_t i = (size_t)blockIdx.x * blockDim.x + threadIdx.x;
    if (i < n) p[i] = 0.0f;
}

// ---------------------------------------------------------------------------
// In-degree via float atomics: deg[dst] += 1
// ---------------------------------------------------------------------------
__global__ void gmm_deg_kernel(const long long* __restrict__ ei,
                               float* __restrict__ deg, int e_cnt) {
    int e = blockIdx.x * blockDim.x + threadIdx.x;
    if (e >= e_cnt) return;
    long long dst = ei[(size_t)e_cnt + e];   // edge_index[1, e]
    atomicAdd(&deg[dst], 1.0f);
}

// ---------------------------------------------------------------------------
// deg = max(deg,1); isd = rsqrt(deg)
// ---------------------------------------------------------------------------
__global__ void gmm_isd_kernel(float* __restrict__ deg,
                               float* __restrict__ isd, int n) {
    int i = blockIdx.x * blockDim.x + threadIdx.x;
    if (i >= n) return;
    float d = fmaxf(deg[i], 1.0f);
    deg[i] = d;
    isd[i] = rsqrtf(d);
}

// ---------------------------------------------------------------------------
// Dense GEMM on the WMMA pipe, full fp32 (V_WMMA_F32_16X16X4_F32).
// Templated on LDW (= weight ld = output ld: 256 for g->xt, 64 for root->xr)
// so every global/LDS access uses a compile-time instruction offset.
//
// Block = 256 threads = 8 waves. Each block owns 128 rows x one 16-col tile.
//   Stage:   64x16 B tile (4 KB) -> LDS via global_load_async_to_lds_b128
//            (ASYNCcnt) ; s_wait_asynccnt 0 ; barrier.
//   Compute: wave w does row tile (blockIdx.x*8 + w): K=64 -> 16 chained
//            16x16x4 WMMAs; A frags as float2 global loads (const offsets),
//            B frags as ds_load from the shared tile.
// Fragment layouts per ISA tables:
//   A 16x4 f32: lane m = lane&15, VGPR0 = K=khalf, VGPR1 = K=khalf+1
//               (khalf = 2*(lane>>4)); B mirrors with lanes indexing N.
//   C/D: VGPR v -> M = v + 8*(lane>>4), N = lane&15.
// ---------------------------------------------------------------------------
template <int LDW>
__global__ __launch_bounds__(256) void gmm_gemm_kernel(
    const float* __restrict__ x, const float* __restrict__ W,
    float* __restrict__ out, int n) {
    __shared__ float sB[64 * 16];           // B tile, row-major [k][ncol]

    const int tid   = threadIdx.x;          // 0..255
    const int lane  = tid & 31;
    const int wid   = tid >> 5;             // 0..7
    const int col0  = blockIdx.y * 16;
    const int rowB  = blockIdx.x * 128;     // 8 row tiles per block

    // ---- async stage of B tile: thread t -> row k = t/4, 4 floats ----
    {
        const int k    = tid >> 2;          // 0..63
        const int part = (tid & 3) << 2;    // 0,4,8,12
        const float* gsrc = W + (size_t)k * LDW + col0 + part;
        unsigned lds_off = (unsigned)(size_t)(lds_f*)(&sB[k * 16 + part]);
        asm volatile("global_load_async_to_lds_b128 %0, %1, off"
                     :: "v"(lds_off), "v"(gsrc) : "memory");
        asm volatile("s_wait_asynccnt 0x0" ::: "memory");
    }
    __syncthreads();

    // ---- compute: one 16x16 tile per wave ----
    const int m     = lane & 15;
    const int khalf = (lane >> 4) << 1;     // 0 or 2
    const int row0  = rowB + wid * 16;

    int arow_i = row0 + m;
    if (arow_i >= n) arow_i = n - 1;        // clamp loads; EXEC all-1 for WMMA
    const float* abase = x + (size_t)arow_i * D_FEAT + khalf;
    const float* bbase = &sB[khalf * 16 + m];

    v8f c = {};
    #pragma unroll
    for (int kk = 0; kk < 16; ++kk) {
        const v2f a = *(const v2f*)(abase + 4 * kk);       // K = 4kk+khalf, +1
        v2f b;
        b.x = bbase[kk * 64];                              // row 4kk+khalf
        b.y = bbase[kk * 64 + 16];                         // row 4kk+khalf+1
        c = __builtin_amdgcn_wmma_f32_16x16x4_f32(
                /*neg_a=*/false, a, /*neg_b=*/false, b,
                /*c_mod=*/(short)0, c, /*reuse_a=*/false, /*reuse_b=*/false);
    }

    const int ncol  = lane & 15;
    const int mbase = row0 + ((lane >> 4) << 3);
    #pragma unroll
    for (int v = 0; v < 8; ++v) {
        const int mrow = mbase + v;
        if (mrow < n) out[(size_t)mrow * LDW + col0 + ncol] = c[v];
    }
}

// ---------------------------------------------------------------------------
// Edge aggregation: one wave32 per edge, 8 waves per block.
// Lane owns features d = lane, lane+32; accumulates over K=4 mixture kernels
// in registers, then ONE global_atomic_add_f32 per feature (64 atomics/edge).
// Whole xt (102 MB) is L2-resident on MI455X (192 MB L2): prefetch the row.
// ---------------------------------------------------------------------------
__global__ __launch_bounds__(256) void gmm_edge_kernel(
    const long long* __restrict__ ei, const float* __restrict__ isd,
    const float* __restrict__ mu, const float* __restrict__ sigma,
    const float* __restrict__ xt, float* __restrict__ agg, int e_cnt) {
    const int lane = threadIdx.x & 31;
    const int wid  = threadIdx.x >> 5;
    const long long e = (long long)blockIdx.x * 8 + wid;
    if (e >= e_cnt) return;

    const long long src = ei[e];
    const long long dst = ei[(size_t)e_cnt + e];

    const float ps = isd[src];
    const float pd = isd[dst];

    const float* xs = xt + (size_t)src * (K_KERN * D_FEAT);
    __builtin_prefetch(xs, 0, 0);                        // global_prefetch_b8
    __builtin_prefetch(xs + 128, 0, 0);

    float gw[K_KERN];
    #pragma unroll
    for (int k = 0; k < K_KERN; ++k) {
        const float d0 = ps - mu[k * 2 + 0];
        const float d1 = pd - mu[k * 2 + 1];
        const float s0 = sigma[k * 2 + 0];
        const float s1 = sigma[k * 2 + 1];
        gw[k] = __expf(-0.5f * (d0 * d0 / (1e-15f + s0 * s0) +
                                d1 * d1 / (1e-15f + s1 * s1)));
    }

    float* ad = agg + (size_t)dst * D_FEAT;
    #pragma unroll
    for (int half = 0; half < 2; ++half) {
        const int d = lane + half * 32;
        float acc = 0.0f;
        #pragma unroll
        for (int k = 0; k < K_KERN; ++k)
            acc += gw[k] * xs[k * D_FEAT + d];
        atomicAdd(&ad[d], acc);
    }
}

// ---------------------------------------------------------------------------
// out = x + relu(agg/deg + x@root + bias)
// ---------------------------------------------------------------------------
__global__ void gmm_final_kernel(const float* __restrict__ x,
                                 const float* __restrict__ agg,
                                 const float* __restrict__ xr,
                                 const float* __restrict__ bias,
                                 const float* __restrict__ degc,
                                 float* __restrict__ out, int n) {
    size_t i = (size_t)blockIdx.x * blockDim.x + threadIdx.x;
    if (i >= (size_t)n * D_FEAT) return;
    const int node = (int)(i >> 6);
    const int d    = (int)(i & 63);
    const float conv = agg[i] / degc[node] + xr[i] + bias[d];
    out[i] = x[i] + fmaxf(conv, 0.0f);
}

// ---------------------------------------------------------------------------
// Host-side launch. Workspace layout (floats):
//   deg[N] | isd[N] | xt[N*256] | xr[N*64] | agg[N*64]   (~155 MB)
// ---------------------------------------------------------------------------
extern "C" void kernel_launch(void* const* d_in, const int* in_sizes, int n_in,
                              void* d_out, int out_size, void* d_ws, size_t ws_size,
                              hipStream_t stream) {
    const float*      x     = (const float*)d_in[0];
    const long long*  ei    = (const long long*)d_in[1];
    const float*      g     = (const float*)d_in[2];
    const float*      mu    = (const float*)d_in[3];
    const float*      sigma = (const float*)d_in[4];
    const float*      root  = (const float*)d_in[5];
    const float*      bias  = (const float*)d_in[6];
    float*            out   = (float*)d_out;

    const int n     = in_sizes[0] / D_FEAT;   // 100000
    const int e_cnt = in_sizes[1] / 2;        // 1600000

    float* deg = (float*)d_ws;
    float* isd = deg + n;
    float* xt  = isd + n;
    float* xr  = xt + (size_t)n * (K_KERN * D_FEAT);
    float* agg = xr + (size_t)n * D_FEAT;

    const size_t zn = (size_t)n;
    const size_t za = (size_t)n * D_FEAT;
    gmm_zero_kernel<<<(unsigned)((zn + 255) / 256), 256, 0, stream>>>(deg, zn);
    gmm_zero_kernel<<<(unsigned)((za + 255) / 256), 256, 0, stream>>>(agg, za);

    gmm_deg_kernel<<<(e_cnt + 255) / 256, 256, 0, stream>>>(ei, deg, e_cnt);
    gmm_isd_kernel<<<(n + 255) / 256, 256, 0, stream>>>(deg, isd, n);

    const unsigned rb = (unsigned)((n + 127) / 128);
    dim3 ggrid(rb, 16);                        // xt = x @ g   (256 cols)
    gmm_gemm_kernel<256><<<ggrid, 256, 0, stream>>>(x, g, xt, n);
    dim3 rgrid(rb, 4);                         // xr = x @ root (64 cols)
    gmm_gemm_kernel<64><<<rgrid, 256, 0, stream>>>(x, root, xr, n);

    gmm_edge_kernel<<<(e_cnt + 7) / 8, 256, 0, stream>>>(ei, isd, mu, sigma, xt, agg, e_cnt);

    gmm_final_kernel<<<(unsigned)(((size_t)n * D_FEAT + 255) / 256), 256, 0, stream>>>(
        x, agg, xr, bias, deg, out, n);
}